// CrossOutLayer_2_62646392980067
// MI455X (gfx1250) — compile-verified
//
#include <hip/hip_runtime.h>
#include <math.h>

typedef __attribute__((ext_vector_type(2))) float v2f;
typedef __attribute__((ext_vector_type(8))) float v8f;

#define DD  128
#define NN1 512
#define NN2 512
#define BBATCH 2

// ---------------------------------------------------------------------------
// Kernel 1: C[M x 128] = A0 @ W0 (+ A1 @ W1) (+ bias per column)
// fp32 WMMA, V_WMMA_F32_16X16X4_F32. One wave per 16x16 C tile.
// Block = 256 threads = 8 waves; wave w owns e-tile w (8 tiles cover N=128).
// grid.x = M/16 row tiles.
// ---------------------------------------------------------------------------
__global__ __launch_bounds__(256) void gemm_wmma_f32(
    const float* __restrict__ A0, const float* __restrict__ A1,
    const float* __restrict__ W0, const float* __restrict__ W1w,
    const float* __restrict__ bias, float* __restrict__ C)
{
    const int lane  = threadIdx.x & 31;
    const int wave  = threadIdx.x >> 5;   // 0..7  -> column tile of C
    const int mtile = blockIdx.x;         // row tile of C
    const int half  = lane >> 4;          // 0 or 1
    const int l16   = lane & 15;

    const int row  = mtile * 16 + l16;    // A-matrix row held by this lane
    const int col  = wave * 16 + l16;     // B/C column held by this lane
    const int koff = half * 2;            // K sub-offset per ISA A/B layout

    v8f acc = {};

    for (int k = 0; k < DD; k += 4) {
        v2f a, b;
        // A 16x4 f32: VGPR0 = K=koff, VGPR1 = K=koff+1 (lane-major M)
        a.x = A0[row * DD + k + koff];
        a.y = A0[row * DD + k + koff + 1];
        // B 4x16 f32: VGPR v holds row K = v + 2*(lane/16), N = lane%16
        b.x = W0[(k + koff) * DD + col];
        b.y = W0[(k + koff + 1) * DD + col];
        acc = __builtin_amdgcn_wmma_f32_16x16x4_f32(
            false, a, false, b, (short)0, acc, false, false);
    }

    if (A1 != nullptr) {
        for (int k = 0; k < DD; k += 4) {
            v2f a, b;
            a.x = A1[row * DD + k + koff];
            a.y = A1[row * DD + k + koff + 1];
            b.x = W1w[(k + koff) * DD + col];
            b.y = W1w[(k + koff + 1) * DD + col];
            acc = __builtin_amdgcn_wmma_f32_16x16x4_f32(
                false, a, false, b, (short)0, acc, false, false);
        }
    }

    const float bv = (bias != nullptr) ? bias[col] : 0.0f;

    // C/D layout: VGPR v -> M = v + 8*(lane/16), N = lane%16
    #pragma unroll
    for (int v = 0; v < 8; ++v) {
        const int r = mtile * 16 + v + half * 8;
        C[r * DD + col] = acc[v] + bv;
    }
}

// ---------------------------------------------------------------------------
// Kernel 2: o[b,n,m] = sum_e gelu(hx[b,n,e] + hy[b,m,e]) * W2[e] + b2
// 256 threads -> one 16(n) x 16(m) output tile. Exact gelu via erff.
// LDS row stride 132 floats: bank stride 4 -> the 16 distinct m rows hit 16
// distinct banks; lanes l and l+16 read identical addresses (broadcast).
// ---------------------------------------------------------------------------
__global__ __launch_bounds__(256) void cross_gelu_out(
    const float* __restrict__ hx, const float* __restrict__ hy,
    const float* __restrict__ W2, const float* __restrict__ b2,
    float* __restrict__ out)
{
    __shared__ float sx[16 * 132];
    __shared__ float sy[16 * 132];
    __shared__ float sw[DD];

    const int mtile = blockIdx.x;   // N2/16
    const int ntile = blockIdx.y;   // N1/16
    const int b     = blockIdx.z;
    const int tid   = threadIdx.x;

    #pragma unroll
    for (int j = 0; j < 8; ++j) {
        const int i = tid + j * 256;
        const int r = i >> 7;       // /128
        const int c = i & 127;
        sx[r * 132 + c] = hx[((size_t)(b * NN1 + ntile * 16 + r)) * DD + c];
        sy[r * 132 + c] = hy[((size_t)(b * NN2 + mtile * 16 + r)) * DD + c];
    }
    if (tid < DD) sw[tid] = W2[tid];
    __syncthreads();

    const int n = tid >> 4;
    const int m = tid & 15;
    const float* px = &sx[n * 132];
    const float* py = &sy[m * 132];

    float acc = 0.0f;
    #pragma unroll 4
    for (int e = 0; e < DD; ++e) {
        const float t = px[e] + py[e];
        const float g = 0.5f * t * (1.0f + erff(t * 0.70710678118654752f));
        acc = fmaf(g, sw[e], acc);
    }

    out[(size_t)b * NN1 * NN2 + (size_t)(ntile * 16 + n) * NN2
        + (size_t)(mtile * 16 + m)] = acc + b2[0];
}

// ---------------------------------------------------------------------------
extern "C" void kernel_launch(void* const* d_in, const int* in_sizes, int n_in,
                              void* d_out, int out_size, void* d_ws, size_t ws_size,
                              hipStream_t stream) {
    const float* x0 = (const float*)d_in[0];   // (B, N1, D)
    const float* x  = (const float*)d_in[1];   // (B, N1, D)
    const float* y  = (const float*)d_in[2];   // (B, N2, D)
    const float* W1 = (const float*)d_in[3];   // (3D, D)
    const float* b1 = (const float*)d_in[4];   // (D,)
    const float* W2 = (const float*)d_in[5];   // (D, 1)
    const float* b2 = (const float*)d_in[6];   // (1,)
    float* out = (float*)d_out;

    const float* Wa = W1;
    const float* Wb = W1 + DD * DD;
    const float* Wc = W1 + 2 * DD * DD;

    float* hx = (float*)d_ws;                          // (B*N1, D) fp32
    float* hy = hx + (size_t)BBATCH * NN1 * DD;        // (B*N2, D) fp32

    // hx = x0@Wa + x@Wb + b1   (M = B*N1 = 1024 rows -> 64 row tiles)
    gemm_wmma_f32<<<dim3(BBATCH * NN1 / 16), 256, 0, stream>>>(
        x0, x, Wa, Wb, b1, hx);

    // hy = y@Wc                (M = B*N2 = 1024 rows)
    gemm_wmma_f32<<<dim3(BBATCH * NN2 / 16), 256, 0, stream>>>(
        y, nullptr, Wc, nullptr, nullptr, hy);

    // fused cross add + exact gelu + weighted e-reduction
    cross_gelu_out<<<dim3(NN2 / 16, NN1 / 16, BBATCH), 256, 0, stream>>>(
        hx, hy, W2, b2, out);
}